// VE_pbc_8624294331192
// MI455X (gfx1250) — compile-verified
//
#include <hip/hip_runtime.h>
#include <math.h>

#define TOTAL_T 1000
#define LOGMIN  -5.2983173665480363   // log(0.005)
#define LOGMAX   2.3025850929940457   // log(10.0)

struct __attribute__((packed, aligned(4))) F3 { float x, y, z; };

// ---------------------------------------------------------------------------
// Main kernel: one workgroup per crystal; streaming per-atom work.
// (Placed first in the TU so the disasm snippet shows it.)
// ---------------------------------------------------------------------------
__global__ __launch_bounds__(256)
void ve_pbc_kernel(const float* __restrict__ x0,
                   const int*   __restrict__ tstep,
                   const float* __restrict__ lattice,
                   const int*   __restrict__ num_atoms,
                   const float* __restrict__ noise,
                   const int*   __restrict__ starts,
                   float* __restrict__ out, int N) {
  const int b = blockIdx.x;
  volatile __shared__ float sL[12];  // 9 used

  // CDNA5 async global->LDS broadcast of this crystal's 3x3 lattice.
  // Wave 0 (fully active, wave32) issues all 32 lanes; lanes 9..31 duplicate
  // k=0 (benign identical-value writes) so EXEC is all-ones for the wave.
  if (threadIdx.x < 32) {
    const int k = (threadIdx.x < 9) ? (int)threadIdx.x : 0;
    unsigned lds_off = (unsigned)(unsigned long long)&sL[k];
    const float* gp  = lattice + (size_t)b * 9 + k;
    asm volatile("global_load_async_to_lds_b32 %0, %1, off\n\t"
                 "s_wait_asynccnt 0x0"
                 :
                 : "v"(lds_off), "v"(gp)
                 : "memory");
  }
  __syncthreads();

  const float L0 = sL[0], L1 = sL[1], L2 = sL[2];
  const float L3 = sL[3], L4 = sL[4], L5 = sL[5];
  const float L6 = sL[6], L7 = sL[7], L8 = sL[8];

  // 3x3 inverse via adjugate in f64 (per-thread; trivial cost, high accuracy)
  const double a00 = L0, a01 = L1, a02 = L2;
  const double a10 = L3, a11 = L4, a12 = L5;
  const double a20 = L6, a21 = L7, a22 = L8;
  const double c00 =  (a11 * a22 - a12 * a21);
  const double c01 = -(a10 * a22 - a12 * a20);
  const double c02 =  (a10 * a21 - a11 * a20);
  const double c10 = -(a01 * a22 - a02 * a21);
  const double c11 =  (a00 * a22 - a02 * a20);
  const double c12 = -(a00 * a21 - a01 * a20);
  const double c20 =  (a01 * a12 - a02 * a11);
  const double c21 = -(a00 * a12 - a02 * a10);
  const double c22 =  (a00 * a11 - a01 * a10);
  const double rd  = 1.0 / (a00 * c00 + a01 * c01 + a02 * c02);
  // inv[i][j] = cofactor(j,i) / det
  const float i00 = (float)(c00 * rd), i01 = (float)(c10 * rd), i02 = (float)(c20 * rd);
  const float i10 = (float)(c01 * rd), i11 = (float)(c11 * rd), i12 = (float)(c21 * rd);
  const float i20 = (float)(c02 * rd), i21 = (float)(c12 * rd), i22 = (float)(c22 * rd);

  const int start = starts[b];
  const int count = num_atoms[b];

  float* __restrict__ out_frac = out;
  float* __restrict__ out_wrap = out + (size_t)3 * N;
  float* __restrict__ out_sig  = out + (size_t)6 * N;

  const float stepf   = (float)((LOGMAX - LOGMIN) / (double)TOTAL_T);
  const float logminf = (float)LOGMIN;

  const F3* __restrict__ x0v = (const F3*)x0;
  const F3* __restrict__ nzv = (const F3*)noise;

  for (int i = threadIdx.x; i < count; i += 256) {
    const int    n    = start + i;
    const size_t idx3 = (size_t)3 * n;

    if (i + 256 < count) {
      __builtin_prefetch(&x0[idx3 + 768], 0, 3);     // next tile of this block
      __builtin_prefetch(&noise[idx3 + 768], 0, 3);
    }

    const F3 a  = x0v[n];    // 12B row -> global_load_b96 candidate
    const F3 nv = nzv[n];
    const int tt = tstep[n];

    const float sg  = expf(fmaf((float)tt, stepf, logminf));
    const float xnx = fmaf(nv.x, sg, a.x);
    const float xny = fmaf(nv.y, sg, a.y);
    const float xnz = fmaf(nv.z, sg, a.z);

    // frac_j = sum_i xn_i * inv[i][j], then mod 1.0 (== v - floor(v))
    float f0 = xnx * i00 + xny * i10 + xnz * i20;
    float f1 = xnx * i01 + xny * i11 + xnz * i21;
    float f2 = xnx * i02 + xny * i12 + xnz * i22;
    f0 -= floorf(f0);
    f1 -= floorf(f1);
    f2 -= floorf(f2);

    // cart_j = sum_i f_i * L[i][j]
    const float cX = f0 * L0 + f1 * L3 + f2 * L6;
    const float cY = f0 * L1 + f1 * L4 + f2 * L7;
    const float cZ = f0 * L2 + f1 * L5 + f2 * L8;

    const float dx = cX - a.x, dy = cY - a.y, dz = cZ - a.z;

    // 27-cell min-image search; cell coords constant-fold on unroll.
    float best = 3.4e38f;
    int   bc   = 0;
#pragma unroll
    for (int c = 0; c < 27; ++c) {
      const float e0 = (float)(c / 9 - 1);
      const float e1 = (float)((c / 3) % 3 - 1);
      const float e2 = (float)(c % 3 - 1);
      const float vx = dx - (e0 * L0 + e1 * L3 + e2 * L6);
      const float vy = dy - (e0 * L1 + e1 * L4 + e2 * L7);
      const float vz = dz - (e0 * L2 + e1 * L5 + e2 * L8);
      const float d2 = vx * vx + vy * vy + vz * vz;
      if (d2 < best) { best = d2; bc = c; }   // strict < keeps first min (argmin)
    }
    const float e0 = (float)(bc / 9 - 1);
    const float e1 = (float)((bc / 3) % 3 - 1);
    const float e2 = (float)(bc % 3 - 1);
    const float wx = dx - (e0 * L0 + e1 * L3 + e2 * L6);
    const float wy = dy - (e0 * L1 + e1 * L4 + e2 * L7);
    const float wz = dz - (e0 * L2 + e1 * L5 + e2 * L8);

    __builtin_nontemporal_store(f0, &out_frac[idx3 + 0]);
    __builtin_nontemporal_store(f1, &out_frac[idx3 + 1]);
    __builtin_nontemporal_store(f2, &out_frac[idx3 + 2]);
    __builtin_nontemporal_store(wx, &out_wrap[idx3 + 0]);
    __builtin_nontemporal_store(wy, &out_wrap[idx3 + 1]);
    __builtin_nontemporal_store(wz, &out_wrap[idx3 + 2]);
    __builtin_nontemporal_store(sg, &out_sig[n]);
  }
}

// ---------------------------------------------------------------------------
// Kernel 1: exclusive prefix sum of num_atoms[B] -> starts[B]  (B <= 4096)
// ---------------------------------------------------------------------------
__global__ __launch_bounds__(1024)
void scan_starts_kernel(const int* __restrict__ num_atoms,
                        int* __restrict__ starts, int B) {
  __shared__ int part[1024];
  const int tid  = threadIdx.x;
  const int base = tid * 4;
  int v0 = (base + 0 < B) ? num_atoms[base + 0] : 0;
  int v1 = (base + 1 < B) ? num_atoms[base + 1] : 0;
  int v2 = (base + 2 < B) ? num_atoms[base + 2] : 0;
  int v3 = (base + 3 < B) ? num_atoms[base + 3] : 0;
  part[tid] = v0 + v1 + v2 + v3;
  __syncthreads();
  // Hillis-Steele inclusive scan over 1024 partials
  for (int off = 1; off < 1024; off <<= 1) {
    int mine = part[tid];
    int add  = (tid >= off) ? part[tid - off] : 0;
    __syncthreads();
    part[tid] = mine + add;
    __syncthreads();
  }
  const int excl = (tid == 0) ? 0 : part[tid - 1];
  if (base + 0 < B) starts[base + 0] = excl;
  if (base + 1 < B) starts[base + 1] = excl + v0;
  if (base + 2 < B) starts[base + 2] = excl + v0 + v1;
  if (base + 3 < B) starts[base + 3] = excl + v0 + v1 + v2;
}

extern "C" void kernel_launch(void* const* d_in, const int* in_sizes, int n_in,
                              void* d_out, int out_size, void* d_ws, size_t ws_size,
                              hipStream_t stream) {
  const float* x0        = (const float*)d_in[0];
  const int*   t         = (const int*)d_in[1];
  const float* lattice   = (const float*)d_in[2];
  const int*   num_atoms = (const int*)d_in[3];
  const float* noise     = (const float*)d_in[4];
  const int B = in_sizes[3];
  const int N = in_sizes[0] / 3;

  int* starts = (int*)d_ws;  // B ints of scratch

  scan_starts_kernel<<<1, 1024, 0, stream>>>(num_atoms, starts, B);
  ve_pbc_kernel<<<B, 256, 0, stream>>>(x0, t, lattice, num_atoms, noise,
                                       starts, (float*)d_out, N);
}